// VectorQuantizer_76854144794891
// MI455X (gfx1250) — compile-verified
//
#include <hip/hip_runtime.h>
#include <hip/hip_bf16.h>
#include <stdint.h>

typedef __attribute__((ext_vector_type(2))) float v2f;
typedef __attribute__((ext_vector_type(8))) float v8f;

#define NUM_EMB   512
#define EMB_DIM   64
#define OUT_ELEMS 4194304   // 64*64*32*32
#define N_ROWS    65536     // 64*32*32

// Low 32 bits of a flat shared-aperture address are the LDS offset
// relative to the workgroup allocation (hardware adds LDS_BASE).
__device__ __forceinline__ uint32_t lds_addr32(const void* p) {
  return (uint32_t)(uintptr_t)p;
}

__global__ __launch_bounds__(256)
void vq_main_kernel(const float* __restrict__ in,     // [64,64,32,32] NCHW
                    const float* __restrict__ emb,    // [512,64]
                    float* __restrict__ d_out,        // [1 + 4194304 + 65536]
                    float* __restrict__ part) {       // [512] block loss partials
  extern __shared__ float smem[];
  float* s_emb   = smem;          // 32768 floats (128 KB codebook)
  float* s_enorm = smem + 32768;  // 512 floats
  float* s_part  = smem + 33280;  // 8 floats (per-wave loss)

  const int tid = threadIdx.x;

  // ---- Stage full codebook into LDS via gfx1250 async global->LDS DMA ----
  // 131072 B / (256 threads * 16 B) = 32 b128 transfers per thread.
  for (int i = 0; i < 32; ++i) {
    const int e4 = i * 256 + tid;                         // float4 index 0..8191
    uint64_t ga  = (uint64_t)(uintptr_t)(emb + e4 * 4);
    uint32_t la  = lds_addr32(s_emb + e4 * 4);
    asm volatile("global_load_async_to_lds_b128 %0, %1, off"
                 :: "v"(la), "v"(ga) : "memory");
  }
  asm volatile("s_wait_asynccnt 0" ::: "memory");
  __syncthreads();

  // ---- Per-code squared norms ----
  for (int k = tid; k < NUM_EMB; k += 256) {
    const float* e = s_emb + k * EMB_DIM;
    float s = 0.f;
    #pragma unroll
    for (int d = 0; d < EMB_DIM; ++d) s += e[d] * e[d];
    s_enorm[k] = s;
  }
  __syncthreads();

  // ---- Per-wave 16-row tile ----
  const int lane = tid & 31;
  const int wave = tid >> 5;
  const int m    = lane & 15;    // row within tile (also code-within-chunk for B/C)
  const int hi   = lane >> 4;    // half-wave select
  const int rowBase = blockIdx.x * 128 + wave * 16;
  const int bIdx = rowBase >> 10;          // batch index (HW=1024 per image)
  const int hw   = rowBase & 1023;         // position within image (16-aligned)
  // x_flat[row][d] = in[bIdx*65536 + d*1024 + hw + m]
  const float* xbase = in + bIdx * 65536 + hw + m;

  // A fragments: 16 steps of 16x4 f32. Element (m,k) lives in lane m+16*(k>>1),
  // vgpr k&1  ->  lane (m,hi) holds dims {4t+2hi, 4t+2hi+1}. Coalesced loads.
  v2f a[16];
  #pragma unroll
  for (int t = 0; t < 16; ++t) {
    const int d0 = 4 * t + 2 * hi;
    a[t].x = xbase[d0 * 1024];
    a[t].y = xbase[d0 * 1024 + 1024];
  }

  float best[8];
  int   bcode[8];
  #pragma unroll
  for (int r = 0; r < 8; ++r) { best[r] = 3.402823466e38f; bcode[r] = 0; }

  // 32 chunks of 16 codes; K=64 accumulated as 16 WMMA f32 16x16x4 steps.
  for (int c = 0; c < 32; ++c) {
    v8f acc = {0.f, 0.f, 0.f, 0.f, 0.f, 0.f, 0.f, 0.f};
    const int code = c * 16 + m;                     // this lane's column / code
    const float* ebase = s_emb + code * EMB_DIM + 2 * hi;
    #pragma unroll
    for (int t = 0; t < 16; ++t) {
      v2f bf = *(const v2f*)(ebase + 4 * t);         // 8B-aligned LDS read
      acc = __builtin_amdgcn_wmma_f32_16x16x4_f32(
          false, a[t], false, bf, (short)0, acc, false, false);
    }
    const float en = s_enorm[code];
    #pragma unroll
    for (int r = 0; r < 8; ++r) {
      // dist = ||e||^2 - 2 x.e  (||x||^2 is row-constant, irrelevant to argmin)
      const float dist = en - 2.0f * acc[r];
      if (dist < best[r]) { best[r] = dist; bcode[r] = code; }  // earliest code wins ties
    }
  }

  // ---- Argmin reduction across the 16 lanes sharing each row group ----
  // C/D layout: vgpr r, lanes 0-15 -> row r; lanes 16-31 -> row r+8.
  #pragma unroll
  for (int r = 0; r < 8; ++r) {
    #pragma unroll
    for (int off = 8; off >= 1; off >>= 1) {   // stays within 16-lane halves
      const float od = __shfl_xor(best[r], off, 32);
      const int   oi = __shfl_xor(bcode[r], off, 32);
      if (od < best[r] || (od == best[r] && oi < bcode[r])) {
        best[r] = od; bcode[r] = oi;
      }
    }
  }

  // Broadcast winning code of row (lane&15) to every lane.
  int myRowCode = 0;
  #pragma unroll
  for (int rr = 0; rr < 8; ++rr) {
    const int vLo = __shfl(bcode[rr], 0, 32);    // row rr
    const int vHi = __shfl(bcode[rr], 16, 32);   // row rr+8
    if (m == rr)     myRowCode = vLo;
    if (m == rr + 8) myRowCode = vHi;
  }

  // ---- Write idx (as f32, reference returns int32 indices) ----
  float* outMain = d_out + 1;
  float* idxOut  = d_out + 1 + OUT_ELEMS;
  if (hi == 0) idxOut[rowBase + m] = (float)myRowCode;

  // ---- quantized_st forward == emb[idx]; write NCHW + loss partial ----
  const float* qrow = s_emb + myRowCode * EMB_DIM;
  float* obase = outMain + bIdx * 65536 + hw + m;
  float lsum = 0.f;
  #pragma unroll
  for (int dd = 0; dd < 32; ++dd) {
    const int d = 2 * dd + hi;          // each (row,d) covered exactly once
    const float q  = qrow[d];
    const float xv = xbase[d * 1024];
    const float df = q - xv;
    lsum += df * df;
    obase[d * 1024] = q;
  }

  // Deterministic loss reduction: wave shfl tree -> LDS -> block partial.
  #pragma unroll
  for (int off = 16; off >= 1; off >>= 1) lsum += __shfl_xor(lsum, off, 32);
  if (lane == 0) s_part[wave] = lsum;
  __syncthreads();
  if (tid == 0) {
    float s = 0.f;
    #pragma unroll
    for (int w = 0; w < 8; ++w) s += s_part[w];
    part[blockIdx.x] = s;
  }
}

__global__ __launch_bounds__(512)
void vq_reduce_kernel(const float* __restrict__ part, float* __restrict__ d_out) {
  __shared__ float s[512];
  const int tid = threadIdx.x;
  s[tid] = part[tid];
  __syncthreads();
  for (int off = 256; off > 0; off >>= 1) {
    if (tid < off) s[tid] += s[tid + off];
    __syncthreads();
  }
  // loss = q_latent + 0.25*e_latent = 1.25 * mean((q-x)^2) over 4194304 elems
  if (tid == 0) d_out[0] = 1.25f * s[0] / 4194304.0f;
}

extern "C" void kernel_launch(void* const* d_in, const int* in_sizes, int n_in,
                              void* d_out, int out_size, void* d_ws, size_t ws_size,
                              hipStream_t stream) {
  const float* in  = (const float*)d_in[0];   // [64,64,32,32] f32
  const float* emb = (const float*)d_in[1];   // [512,64] f32
  float* out  = (float*)d_out;
  float* part = (float*)d_ws;                 // 512 floats of scratch

  const size_t shmem = (size_t)(32768 + 512 + 8) * sizeof(float);  // ~130 KB of 320 KB LDS
  (void)hipFuncSetAttribute((const void*)vq_main_kernel,
                            hipFuncAttributeMaxDynamicSharedMemorySize, (int)shmem);

  vq_main_kernel<<<512, 256, shmem, stream>>>(in, emb, out, part);
  vq_reduce_kernel<<<1, 512, 0, stream>>>(part, out);
}